// SingleLayerLstm_51694226374665
// MI455X (gfx1250) — compile-verified
//
#include <hip/hip_runtime.h>
#include <math.h>

// ---------------------------------------------------------------------------
// Single-layer LSTM, fp32, CDNA5 (gfx1250) using V_WMMA_F32_16X16X4_F32.
// z_t = [X_t ; h_{t-1}] @ [Wf ; Wr] + bf   (fused concatenated-K GEMM, K=1024)
// Recurrent state is carried through the Y / C output arrays themselves.
// ---------------------------------------------------------------------------

typedef float v2f __attribute__((ext_vector_type(2)));
typedef float v8f __attribute__((ext_vector_type(8)));

#define T_DIM  512
#define B_DIM  128
#define NI_DIM 512
#define D_DIM  512
#define G4     (4 * D_DIM)   // 2048 columns of z

__device__ __forceinline__ float sigmoid_fast(float x) {
    return 1.0f / (1.0f + __expf(-x));
}
__device__ __forceinline__ float tanh_fast(float x) {
    // 1 - 2/(e^{2x}+1): saturates correctly at +/-1 for large |x|
    float e = __expf(2.0f * x);
    return 1.0f - 2.0f / (e + 1.0f);
}

// One timestep: grid (D/64, B/32), block 128 threads (4 waves).
// Wave w computes the 32x16 y/c tile at rows b0..b0+31, cols d0+16w..d0+16w+15,
// accumulating all four gate blocks of z for those columns.
__global__ __launch_bounds__(128)
void lstm_step_kernel(const float* __restrict__ Xt,     // B x NI  (timestep t)
                      const float* __restrict__ Wf,     // NI x 4D
                      const float* __restrict__ Wr,     // D  x 4D
                      const float* __restrict__ bf,     // 4D
                      const float* __restrict__ hPrev,  // B x D
                      const float* __restrict__ cPrev,  // B x D
                      const float* __restrict__ it,     // B   (mask row at t)
                      float* __restrict__ Yt,           // B x D
                      float* __restrict__ Ct,           // B x D
                      float* __restrict__ dOut)         // B x D or nullptr
{
    const int lane = threadIdx.x & 31;
    const int wave = threadIdx.x >> 5;                  // 0..3
    const int d0   = blockIdx.x * 64 + wave * 16;       // d-column tile start
    const int b0   = blockIdx.y * 32;                   // batch tile start

    const int rowA = lane & 15;                         // A-fragment row
    const int kSel = (lane >> 4) << 1;                  // 0 or 2 (K pair select)
    const int col  = lane & 15;                         // B/C-fragment column

    // --- accumulators: [m-subtile][gate], initialized with the bias ---------
    v8f acc[2][4];
#pragma unroll
    for (int g = 0; g < 4; ++g) {
        float bv = bf[g * D_DIM + d0 + col];
#pragma unroll
        for (int m = 0; m < 2; ++m)
#pragma unroll
            for (int r = 0; r < 8; ++r) acc[m][g][r] = bv;
    }

    // --- Phase 1: X_t @ Wf  (K = NI) ---------------------------------------
    {
        const float* a0p = Xt + (size_t)(b0 + rowA) * NI_DIM + kSel;
        const float* a1p = Xt + (size_t)(b0 + 16 + rowA) * NI_DIM + kSel;
#pragma unroll 1
        for (int k0 = 0; k0 < NI_DIM; k0 += 4) {
            v2f a0 = *(const v2f*)(a0p + k0);
            v2f a1 = *(const v2f*)(a1p + k0);
            const float* w0 = Wf + (size_t)(k0 + kSel) * G4;
            const float* w1 = w0 + G4;
#pragma unroll
            for (int g = 0; g < 4; ++g) {
                const int c = g * D_DIM + d0 + col;
                v2f b; b.x = w0[c]; b.y = w1[c];
                acc[0][g] = __builtin_amdgcn_wmma_f32_16x16x4_f32(
                    false, a0, false, b, (short)0, acc[0][g], false, false);
                acc[1][g] = __builtin_amdgcn_wmma_f32_16x16x4_f32(
                    false, a1, false, b, (short)0, acc[1][g], false, false);
            }
        }
    }

    // --- Phase 2: h_{t-1} @ Wr  (K = D) ------------------------------------
    {
        const float* a0p = hPrev + (size_t)(b0 + rowA) * D_DIM + kSel;
        const float* a1p = hPrev + (size_t)(b0 + 16 + rowA) * D_DIM + kSel;
#pragma unroll 1
        for (int k0 = 0; k0 < D_DIM; k0 += 4) {
            v2f a0 = *(const v2f*)(a0p + k0);
            v2f a1 = *(const v2f*)(a1p + k0);
            const float* w0 = Wr + (size_t)(k0 + kSel) * G4;
            const float* w1 = w0 + G4;
#pragma unroll
            for (int g = 0; g < 4; ++g) {
                const int c = g * D_DIM + d0 + col;
                v2f b; b.x = w0[c]; b.y = w1[c];
                acc[0][g] = __builtin_amdgcn_wmma_f32_16x16x4_f32(
                    false, a0, false, b, (short)0, acc[0][g], false, false);
                acc[1][g] = __builtin_amdgcn_wmma_f32_16x16x4_f32(
                    false, a1, false, b, (short)0, acc[1][g], false, false);
            }
        }
    }

    // --- Epilogue: gate math + cell update, all in registers ----------------
    // C/D fragment layout: VGPR r, lanes 0-15 -> row r; lanes 16-31 -> row r+8.
    const int rowOfs = (lane >> 4) << 3;
#pragma unroll
    for (int m = 0; m < 2; ++m) {
#pragma unroll
        for (int r = 0; r < 8; ++r) {
            const int b = b0 + m * 16 + rowOfs + r;
            const int d = d0 + col;
            const size_t idx = (size_t)b * D_DIM + d;

            float cin = tanh_fast(acc[m][0][r]);
            float ig  = sigmoid_fast(acc[m][1][r]);
            float fg  = sigmoid_fast(acc[m][2][r]);
            float og  = sigmoid_fast(acc[m][3][r]);

            float cp = cPrev[idx];
            float mk = it[b];
            float cn = mk * (cp * fg + cin * ig) + (1.0f - mk) * cp;
            float y  = mk * (og * tanh_fast(cn));

            Yt[idx] = y;
            Ct[idx] = cn;
            if (dOut) dOut[idx] = cn;   // final carry (last timestep only)
        }
    }
}

extern "C" void kernel_launch(void* const* d_in, const int* in_sizes, int n_in,
                              void* d_out, int out_size, void* d_ws, size_t ws_size,
                              hipStream_t stream) {
    const float* X  = (const float*)d_in[0];   // T x B x NI
    const float* Wf = (const float*)d_in[1];   // NI x 4D
    const float* Wr = (const float*)d_in[2];   // D x 4D
    const float* bf = (const float*)d_in[3];   // 4D
    const float* iM = (const float*)d_in[4];   // T x B
    const float* h0 = (const float*)d_in[5];   // B x D
    const float* c0 = (const float*)d_in[6];   // B x D

    float* Y = (float*)d_out;                              // T*B*D
    float* C = Y + (size_t)T_DIM * B_DIM * D_DIM;          // T*B*D
    float* d = C + (size_t)T_DIM * B_DIM * D_DIM;          // B*D (final c)

    dim3 grid(D_DIM / 64, B_DIM / 32);   // (8, 4)
    dim3 block(128);                     // 4 wave32 per block

    for (int t = 0; t < T_DIM; ++t) {
        const float* Xt = X + (size_t)t * B_DIM * NI_DIM;
        const float* hP = (t == 0) ? h0 : (Y + (size_t)(t - 1) * B_DIM * D_DIM);
        const float* cP = (t == 0) ? c0 : (C + (size_t)(t - 1) * B_DIM * D_DIM);
        const float* it = iM + (size_t)t * B_DIM;
        float* Yt = Y + (size_t)t * B_DIM * D_DIM;
        float* Ct = C + (size_t)t * B_DIM * D_DIM;
        float* dO = (t == T_DIM - 1) ? d : nullptr;

        lstm_step_kernel<<<grid, block, 0, stream>>>(Xt, Wf, Wr, bf, hP, cP, it,
                                                     Yt, Ct, dO);
    }
}